// CustomPunitiveLoss_82102594830633
// MI455X (gfx1250) — compile-verified
//
#include <hip/hip_runtime.h>
#include <hip/hip_bf16.h>
#include <math.h>

// ---------------------------------------------------------------------------
// CustomPunitiveLoss for MI455X (gfx1250, wave32).
//   loss_row = -log p_t + 0.1 * (C - 2 + S2/S1^2 - (1 - p_t)^2)
//   with m = rowmax, S1 = sum exp(x-m), S2 = sum exp(2(x-m)), p_t = exp(x_t-m)/S1
// Bandwidth-bound: 1.05 GB @ 23.3 TB/s => ~45 us floor. Pass 2 re-read hits L2.
// ---------------------------------------------------------------------------

typedef __attribute__((ext_vector_type(2))) float v2f;
typedef __attribute__((ext_vector_type(8))) float v8f;

// Exact sum of one value from each of the 32 lanes of the executing wave,
// using V_WMMA_F32_16X16X4_F32 (D = A x B, B = ones(4x16)).
// Documented A layout (16x4 f32): lanes 0-15: VGPR0=K0, VGPR1=K1 (M=lane);
// lanes 16-31: VGPR0=K2, VGPR1=K3 (M=lane-16). With a[1]=0:
//   rowsum A[m] = v_m + v_{m+16}  ->  D[m][n] = v_m + v_{m+16} for all n.
// Documented D layout: lane n holds D[0..7][n], lane n+16 holds D[8..15][n],
// so (sum of my 8 D regs) + (xor-16 partner's) = sum over all 32 lanes.
// Must be executed with EXEC all-ones for the wave (whole-wave branch only).
__device__ __forceinline__ float wave32_sum_wmma(float v) {
  v2f a; a[0] = v;    a[1] = 0.0f;
  v2f b; b[0] = 1.0f; b[1] = 1.0f;
  v8f c = {};
  v8f d = __builtin_amdgcn_wmma_f32_16x16x4_f32(
      /*neg_a=*/false, a, /*neg_b=*/false, b,
      /*c_mod=*/(short)0, c, /*reuse_a=*/false, /*reuse_b=*/false);
  float s = ((d[0] + d[1]) + (d[2] + d[3])) + ((d[4] + d[5]) + (d[6] + d[7]));
  s += __shfl_xor(s, 16, 32);
  return s;
}

__global__ __launch_bounds__(256) void punitive_row_kernel(
    const float* __restrict__ input, const int* __restrict__ target,
    float* __restrict__ row_loss, int C) {
  const int row = blockIdx.x;
  const int tid = threadIdx.x;
  const float* __restrict__ rp = input + (size_t)row * (size_t)C;

  __shared__ float redmax[256];
  __shared__ float sA[256];
  __shared__ float sB[256];
  __shared__ float sh_xt;

  if (tid == 0) {
    const int t = target[row];
    sh_xt = rp[t];
  }

  // Vector path only when the row base/length are 16B-friendly (true here: C=32000).
  const int C4 = ((C & 3) == 0) ? (C >> 2) : 0;
  const float4* __restrict__ rp4 = (const float4*)rp;

  // ---- Pass 1: row max (streams the row from HBM, b128 loads) ----
  float m = -INFINITY;
  for (int i = tid; i < C4; i += 256) {
    const float4 v = rp4[i];
    m = fmaxf(m, fmaxf(fmaxf(v.x, v.y), fmaxf(v.z, v.w)));
  }
  for (int i = (C4 << 2) + tid; i < C; i += 256) m = fmaxf(m, rp[i]);

  redmax[tid] = m;
  __syncthreads();
#pragma unroll
  for (int s = 128; s > 0; s >>= 1) {
    if (tid < s) redmax[tid] = fmaxf(redmax[tid], redmax[tid + s]);
    __syncthreads();
  }
  const float rowmax = redmax[0];

  // ---- Pass 2: S1, S2 (row is L2-resident from pass 1; hardware v_exp_f32) ----
  float s1 = 0.0f, s2 = 0.0f;
  for (int i = tid; i < C4; i += 256) {
    const float4 v = rp4[i];
    const float e0 = __expf(v.x - rowmax);
    const float e1 = __expf(v.y - rowmax);
    const float e2 = __expf(v.z - rowmax);
    const float e3 = __expf(v.w - rowmax);
    s1 += (e0 + e1) + (e2 + e3);
    s2 += (e0 * e0 + e1 * e1) + (e2 * e2 + e3 * e3);
  }
  for (int i = (C4 << 2) + tid; i < C; i += 256) {
    const float e = __expf(rp[i] - rowmax);
    s1 += e;
    s2 += e * e;
  }

  sA[tid] = s1;
  sB[tid] = s2;
  __syncthreads();

  // ---- 256 -> 32 via LDS, final 32 -> 1 via WMMA (wave 0 only, EXEC all-1s) ----
  if (tid < 32) {
    float a1 = 0.0f, a2 = 0.0f;
#pragma unroll
    for (int k = 0; k < 256; k += 32) {
      a1 += sA[tid + k];
      a2 += sB[tid + k];
    }
    const float S1 = wave32_sum_wmma(a1);
    const float S2 = wave32_sum_wmma(a2);
    if (tid == 0) {
      const float xt     = sh_xt;
      const float lse    = __logf(S1);
      const float logpt  = xt - rowmax - lse;
      const float pt     = __expf(logpt);
      const float sum_p2 = S2 / (S1 * S1);
      const float omp    = 1.0f - pt;
      const float punish = (float)C - 2.0f + sum_p2 - omp * omp;
      row_loss[row] = -logpt + 0.1f * punish;
    }
  }
}

// Deterministic mean of N per-row losses -> out[0]. Single block; fixed order.
__global__ __launch_bounds__(256) void mean_reduce_kernel(
    const float* __restrict__ vals, float* __restrict__ out, int N) {
  const int tid = threadIdx.x;
  float s = 0.0f;
  for (int i = tid; i < N; i += 256) s += vals[i];

  __shared__ float sA[256];
  sA[tid] = s;
  __syncthreads();

  if (tid < 32) {
    float a = 0.0f;
#pragma unroll
    for (int k = 0; k < 256; k += 32) a += sA[tid + k];
    const float tot = wave32_sum_wmma(a);
    if (tid == 0) out[0] = tot / (float)N;
  }
}

extern "C" void kernel_launch(void* const* d_in, const int* in_sizes, int n_in,
                              void* d_out, int out_size, void* d_ws, size_t ws_size,
                              hipStream_t stream) {
  const float* input  = (const float*)d_in[0];   // [N, C] fp32 logits
  const int*   target = (const int*)d_in[1];     // [N] labels
  float*       out    = (float*)d_out;           // scalar

  const int N = in_sizes[1];
  const int C = in_sizes[0] / N;

  float* row_loss = (float*)d_ws;                // N floats of scratch

  punitive_row_kernel<<<N, 256, 0, stream>>>(input, target, row_loss, C);
  mean_reduce_kernel<<<1, 256, 0, stream>>>(row_loss, out, N);
}